// BasicDeformableConv2D_55456617725981
// MI455X (gfx1250) — compile-verified
//
#include <hip/hip_runtime.h>

typedef __attribute__((ext_vector_type(2))) float v2f;
typedef __attribute__((ext_vector_type(8))) float v8f;

#define B_   8
#define C_   64
#define H_   128
#define W_   128
#define O_   64
#define HW_  (H_ * W_)
#define CK_  576          // C * 9
#define NP_  16           // pixels per block tile

__device__ __forceinline__ v8f wmma4(v2f a, v2f b, v8f c) {
  // D = A(16x4 f32) * B(4x16 f32) + C(16x16 f32)
  return __builtin_amdgcn_wmma_f32_16x16x4_f32(
      /*neg_a=*/false, a, /*neg_b=*/false, b,
      /*c_mod=*/(short)0, c, /*reuse_a=*/false, /*reuse_b=*/false);
}

__global__ __launch_bounds__(128) void dcnv2_fused_kernel(
    const float* __restrict__ x,      // [B,C,H,W]
    const float* __restrict__ w_off,  // [18,C,3,3]
    const float* __restrict__ b_off,  // [18]
    const float* __restrict__ w_mask, // [9,C,3,3]
    const float* __restrict__ b_mask, // [9]
    const float* __restrict__ w_dcn,  // [O,C,3,3] == [O][CK]
    float* __restrict__ out)          // [B,O,H,W]
{
  __shared__ float cols[CK_ * NP_];   // 36 KB, reused for both im2cols
  __shared__ float offm[27 * NP_];    // offset(18) + sigmoid(mask)(9)
  __shared__ int4   s_idx[9 * NP_];   // 4 clamped corner indices per (tap,pix)
  __shared__ float4 s_wt [9 * NP_];   // 4 bilinear weights (mask+validity folded)

  const int tid  = threadIdx.x;
  const int lane = tid & 31;
  const int wave = tid >> 5;
  const int nlan = lane & 15;               // M (for A) / N (for B,C)
  const int koff = (lane >> 4) << 1;        // K sub-offset: lanes 0-15 -> 0, 16-31 -> 2

  const int pix0 = blockIdx.x * NP_;
  const int b    = pix0 >> 14;              // / (H*W)
  const int rem  = pix0 & (HW_ - 1);
  const int h    = rem >> 7;                // / W
  const int w0   = rem & (W_ - 1);

  const float* xb = x + (size_t)b * C_ * HW_;

  // ---- Phase 1: regular im2col (3x3, pad=1) for the offset/mask conv ----
  for (int t = tid; t < CK_ * NP_; t += 128) {
    int p  = t & (NP_ - 1);
    int ck = t >> 4;
    int c  = ck / 9, k = ck - c * 9;
    int ky = k / 3,  kx = k - ky * 3;
    int yy = h + ky - 1;
    int xx = w0 + p + kx - 1;
    float v = 0.f;
    if (yy >= 0 && yy < H_ && xx >= 0 && xx < W_)
      v = xb[c * HW_ + yy * W_ + xx];
    cols[ck * NP_ + p] = v;
  }
  __syncthreads();

  // ---- Phase 2: WMMA GEMM -> 27 offset/mask channels (waves 0,1) ----
  if (wave < 2) {
    int m = wave * 16 + nlan;   // A row this lane feeds (rows 27..31 are dead)
    const float* Arow = (m < 18) ? (w_off + m * CK_)
                       : (m < 27) ? (w_mask + (m - 18) * CK_)
                       : w_off;   // harmless filler; rows 27..31 never stored
    v8f acc = {0.f, 0.f, 0.f, 0.f, 0.f, 0.f, 0.f, 0.f};
#pragma unroll 4
    for (int kb = 0; kb < CK_; kb += 4) {
      v2f a  = *(const v2f*)(Arow + kb + koff);
      v2f bb;
      bb.x = cols[(kb + koff)     * NP_ + nlan];
      bb.y = cols[(kb + koff + 1) * NP_ + nlan];
      acc = wmma4(a, bb, acc);
    }
    int m0 = wave * 16 + ((lane >= 16) ? 8 : 0);
#pragma unroll
    for (int r = 0; r < 8; r++) {
      int mm = m0 + r;
      if (mm < 27) {
        float v = acc[r];
        if (mm < 18) {
          v += b_off[mm];
        } else {
          v += b_mask[mm - 18];
          v = __builtin_amdgcn_rcpf(1.f + __expf(-v));   // fast sigmoid
        }
        offm[mm * NP_ + nlan] = v;
      }
    }
  } else {
    // Waves 2,3 are idle during the small GEMM: prefetch the w_dcn rows they
    // will consume in phase 5 (global_prefetch_b8), warming WGP$/L2.
    const float* Arow = w_dcn + (size_t)(wave * 16 + nlan) * CK_;
#pragma unroll
    for (int pb = 0; pb < CK_; pb += 32)   // 128B stride covers the 2304B row
      __builtin_prefetch(Arow + pb, 0, 3);
  }
  __syncthreads();

  // ---- Phase 3: bilinear corner indices/weights per (tap, pixel) ----
  for (int e = tid; e < 9 * NP_; e += 128) {
    int k = e >> 4, p = e & (NP_ - 1);
    int ky = k / 3, kx = k - ky * 3;
    float dy  = offm[(2 * k)     * NP_ + p];
    float dx  = offm[(2 * k + 1) * NP_ + p];
    float msk = offm[(18 + k)    * NP_ + p];
    float py = (float)(h  - 1 + ky)      + dy;
    float px = (float)(w0 + p - 1 + kx)  + dx;
    float fy = floorf(py), fx = floorf(px);
    float wy1 = py - fy, wx1 = px - fx;
    int y0 = (int)fy, x0i = (int)fx;
    int y1 = y0 + 1,  x1i = x0i + 1;
    float vy0 = (y0  >= 0 && y0  < H_) ? 1.f : 0.f;
    float vy1 = (y1  >= 0 && y1  < H_) ? 1.f : 0.f;
    float vx0 = (x0i >= 0 && x0i < W_) ? 1.f : 0.f;
    float vx1 = (x1i >= 0 && x1i < W_) ? 1.f : 0.f;
    int cy0 = min(max(y0, 0), H_ - 1),  cy1 = min(max(y1, 0), H_ - 1);
    int cx0 = min(max(x0i, 0), W_ - 1), cx1 = min(max(x1i, 0), W_ - 1);
    s_idx[e] = make_int4(cy0 * W_ + cx0, cy0 * W_ + cx1,
                         cy1 * W_ + cx0, cy1 * W_ + cx1);
    s_wt[e]  = make_float4((1.f - wy1) * (1.f - wx1) * msk * vy0 * vx0,
                           (1.f - wy1) * wx1         * msk * vy0 * vx1,
                           wy1 * (1.f - wx1)         * msk * vy1 * vx0,
                           wy1 * wx1                 * msk * vy1 * vx1);
  }
  __syncthreads();

  // ---- Phase 4: deformable im2col (reuse `cols`); 2x ds_load_b128 + 4 gathers
  for (int t = tid; t < CK_ * NP_; t += 128) {
    int p  = t & (NP_ - 1);
    int ck = t >> 4;
    int c  = ck / 9, k = ck - c * 9;
    int e  = k * NP_ + p;
    const float* xc = xb + c * HW_;
    int4   ii = s_idx[e];
    float4 aa = s_wt[e];
    float v = aa.x * xc[ii.x] + aa.y * xc[ii.y]
            + aa.z * xc[ii.z] + aa.w * xc[ii.w];
    cols[ck * NP_ + p] = v;
  }
  __syncthreads();

  // ---- Phase 5: main WMMA GEMM, 4 waves x 16 output channels ----
  {
    const float* Arow = w_dcn + (size_t)(wave * 16 + nlan) * CK_;
    v8f acc = {0.f, 0.f, 0.f, 0.f, 0.f, 0.f, 0.f, 0.f};
#pragma unroll 4
    for (int kb = 0; kb < CK_; kb += 4) {
      v2f a  = *(const v2f*)(Arow + kb + koff);
      v2f bb;
      bb.x = cols[(kb + koff)     * NP_ + nlan];
      bb.y = cols[(kb + koff + 1) * NP_ + nlan];
      acc = wmma4(a, bb, acc);
    }
    float* outp = out + (size_t)b * O_ * HW_ + (size_t)h * W_ + w0;
    int o0 = wave * 16 + ((lane >= 16) ? 8 : 0);
#pragma unroll
    for (int r = 0; r < 8; r++) {
      outp[(size_t)(o0 + r) * HW_ + nlan] = acc[r];
    }
  }
}

extern "C" void kernel_launch(void* const* d_in, const int* in_sizes, int n_in,
                              void* d_out, int out_size, void* d_ws, size_t ws_size,
                              hipStream_t stream) {
  const float* x      = (const float*)d_in[0];
  const float* w_off  = (const float*)d_in[1];
  const float* b_off  = (const float*)d_in[2];
  const float* w_mask = (const float*)d_in[3];
  const float* b_mask = (const float*)d_in[4];
  const float* w_dcn  = (const float*)d_in[5];
  float* out = (float*)d_out;

  dim3 grid((B_ * HW_) / NP_);   // 8192 tiles of 16 pixels
  dcnv2_fused_kernel<<<grid, 128, 0, stream>>>(x, w_off, b_off, w_mask, b_mask,
                                               w_dcn, out);
  (void)d_ws; (void)ws_size; (void)in_sizes; (void)n_in; (void)out_size;
}